// AttentionBlock_79336635892670
// MI455X (gfx1250) — compile-verified
//
#include <hip/hip_runtime.h>
#include <hip/hip_bf16.h>

typedef __attribute__((ext_vector_type(2))) float v2f;
typedef __attribute__((ext_vector_type(8))) float v8f;

#define NN      8192
#define IN_F    256
#define OUT_F   128
#define N_E     262144
#define NEG_SLOPE 0.2f
#define NORM_EPS  1e-12f

#define LDS_STRIDE 258                 // 256 + 2 pad: keeps b64 8B-aligned, spreads banks
#define SMEM_BYTES (OUT_F * LDS_STRIDE * 4)   // 128 * 258 * 4 = 132096 B (< 320KB WGP LDS)

// ---------------------------------------------------------------------------
// 0) zero colnorm2 (8192) and d_out (8192*128) — out is accumulated via atomics
// ---------------------------------------------------------------------------
__global__ void k_init(float* __restrict__ colnorm2, float* __restrict__ out) {
    int tid = blockIdx.x * blockDim.x + threadIdx.x;
    int total = NN + NN * OUT_F;
    for (int i = tid; i < total; i += gridDim.x * blockDim.x) {
        if (i < NN) colnorm2[i] = 0.0f;
        else        out[i - NN] = 0.0f;
    }
}

// ---------------------------------------------------------------------------
// 1) HW = features[8192x256] @ W[256x128], f32 WMMA 16x16x4.
//    One wave per 16x128 output STRIP (8 accumulator tiles): each feature
//    element is loaded from global exactly once (A traffic = 8 MB minimum).
//    W is staged transposed into LDS once per workgroup; B-frags are single
//    ds_load_b64 ops.
//    A frag (16x4 f32): lanes 0-15 hold K=0,1 ; lanes 16-31 hold K=2,3.
//    B frag (4x16 f32): mirrored, N = lane&15.
//    C/D (16x16 f32): 8 VGPRs; lanes<16 -> M=r, lanes>=16 -> M=r+8; N=lane&15.
// ---------------------------------------------------------------------------
__global__ void k_gemm_hw(const float* __restrict__ feat,
                          const float* __restrict__ W,
                          float* __restrict__ HW) {
    extern __shared__ float sWt[];     // [OUT_F][LDS_STRIDE] = W transposed, padded

    // ---- stage W (coalesced global reads, padded transposed LDS writes) ----
    for (int idx = threadIdx.x; idx < IN_F * OUT_F; idx += blockDim.x) {
        int k = idx >> 7;              // row of W   (0..255)
        int c = idx & (OUT_F - 1);     // col of W   (0..127)
        sWt[c * LDS_STRIDE + k] = W[idx];
    }
    __syncthreads();

    int gw    = (blockIdx.x * blockDim.x + threadIdx.x) >> 5;  // global wave id
    int lane  = threadIdx.x & 31;
    int strip = gw;                    // 512 strips of 16 rows
    if (strip >= NN / 16) return;      // wave-uniform: EXEC stays all-ones

    int half = lane >> 4;              // 0: K pair (0,1); 1: K pair (2,3)
    int l15  = lane & 15;
    int row  = strip * 16 + l15;

    const float* arow = feat + (long)row * IN_F + half * 2;    // A[row][k + half*2]

    v8f acc[8];
#pragma unroll
    for (int tn = 0; tn < 8; ++tn) acc[tn] = (v8f){};

#pragma unroll 4
    for (int k = 0; k < IN_F; k += 4) {
        v2f a; a.x = arow[k]; a.y = arow[k + 1];
        __builtin_prefetch(arow + k + 64, 0, 1);   // global_prefetch_b8, ~256B ahead
#pragma unroll
        for (int tn = 0; tn < 8; ++tn) {
            int col = tn * 16 + l15;
            // b.x = W[k+half*2][col], b.y = W[k+half*2+1][col]  (one ds_load_b64)
            v2f b = *(const v2f*)(sWt + col * LDS_STRIDE + k + half * 2);
            acc[tn] = __builtin_amdgcn_wmma_f32_16x16x4_f32(
                /*neg_a=*/false, a, /*neg_b=*/false, b,
                /*c_mod=*/(short)0, acc[tn], /*reuse_a=*/false, /*reuse_b=*/false);
        }
    }

    // ---- store 16x128 strip ----
    float* dbase = HW + (long)(strip * 16 + half * 8) * OUT_F + l15;
#pragma unroll
    for (int tn = 0; tn < 8; ++tn) {
#pragma unroll
        for (int r = 0; r < 8; ++r)
            dbase[(long)r * OUT_F + tn * 16] = acc[tn][r];
    }
}

// ---------------------------------------------------------------------------
// 2) s[r] = HW[r] . a[0:128], t[r] = HW[r] . a[128:256] — one wave per row
// ---------------------------------------------------------------------------
__global__ void k_st(const float* __restrict__ HW,
                     const float* __restrict__ avec,
                     float* __restrict__ s, float* __restrict__ t) {
    int row  = (blockIdx.x * blockDim.x + threadIdx.x) >> 5;
    int lane = threadIdx.x & 31;
    if (row >= NN) return;
    const float* h = HW + (long)row * OUT_F;
    float sa = 0.0f, ta = 0.0f;
#pragma unroll
    for (int c = lane; c < OUT_F; c += 32) {
        float hv = h[c];
        sa += hv * avec[c];
        ta += hv * avec[OUT_F + c];
    }
#pragma unroll
    for (int off = 16; off > 0; off >>= 1) {
        sa += __shfl_xor(sa, off, 32);
        ta += __shfl_xor(ta, off, 32);
    }
    if (lane == 0) { s[row] = sa; t[row] = ta; }
}

// ---------------------------------------------------------------------------
// 3) per-edge leaky-relu score + column sum-of-squares (atomic)
// ---------------------------------------------------------------------------
__global__ void k_edge_score(const int* __restrict__ eidx,
                             const float* __restrict__ s,
                             const float* __restrict__ t,
                             float* __restrict__ lr_e,
                             float* __restrict__ colnorm2) {
    int e = blockIdx.x * blockDim.x + threadIdx.x;
    if (e >= N_E) return;
    int i = eidx[e];          // row (source)
    int j = eidx[N_E + e];    // column (target)
    float v  = s[i] + t[j];
    float lr = (v > 0.0f) ? v : NEG_SLOPE * v;
    lr_e[e] = lr;
    atomicAdd(&colnorm2[j], lr * lr);
}

// ---------------------------------------------------------------------------
// 4) colnorm[j] = max(sqrt(sum e^2), eps)   (in place)
// ---------------------------------------------------------------------------
__global__ void k_norm(float* __restrict__ colnorm2) {
    int j = blockIdx.x * blockDim.x + threadIdx.x;
    if (j < NN) colnorm2[j] = fmaxf(sqrtf(colnorm2[j]), NORM_EPS);
}

// ---------------------------------------------------------------------------
// 5) out[i,:] += (lr_e / colnorm[j]) * HW[j,:] — one wave per edge, 4 f32/lane
// ---------------------------------------------------------------------------
__global__ void k_spmm(const int* __restrict__ eidx,
                       const float* __restrict__ HW,
                       const float* __restrict__ lr_e,
                       const float* __restrict__ colnorm,
                       float* __restrict__ out) {
    int e    = (blockIdx.x * blockDim.x + threadIdx.x) >> 5;
    int lane = threadIdx.x & 31;
    if (e >= N_E) return;
    int i = eidx[e];
    int j = eidx[N_E + e];
    float coeff = lr_e[e] / colnorm[j];
    const float4 h = *(const float4*)(HW + (long)j * OUT_F + lane * 4);
    float* o = out + (long)i * OUT_F + lane * 4;
    atomicAdd(o + 0, coeff * h.x);
    atomicAdd(o + 1, coeff * h.y);
    atomicAdd(o + 2, coeff * h.z);
    atomicAdd(o + 3, coeff * h.w);
}

// ---------------------------------------------------------------------------
extern "C" void kernel_launch(void* const* d_in, const int* in_sizes, int n_in,
                              void* d_out, int out_size, void* d_ws, size_t ws_size,
                              hipStream_t stream) {
    const float* feat = (const float*)d_in[0];   // [8192,256]
    const float* W    = (const float*)d_in[1];   // [256,128]
    const float* avec = (const float*)d_in[2];   // [256,1]
    const int*   eidx = (const int*)d_in[3];     // [2,262144]
    float* out = (float*)d_out;                  // [8192,128]

    // workspace layout (f32): HW | s | t | colnorm2 | lr_e  ~5.1 MB
    float* ws = (float*)d_ws;
    float* HW       = ws;                         // 8192*128
    float* s        = HW + (long)NN * OUT_F;      // 8192
    float* t        = s + NN;                     // 8192
    float* colnorm2 = t + NN;                     // 8192
    float* lr_e     = colnorm2 + NN;              // 262144

    // 0) zero accumulators
    k_init<<<1024, 256, 0, stream>>>(colnorm2, out);

    // 1) HW = feat @ W : 512 strip-waves, 8 waves/block, W staged in LDS
    k_gemm_hw<<<(NN / 16) / 8, 256, SMEM_BYTES, stream>>>(feat, W, HW);

    // 2) s,t : 8192 row-waves, 8 waves/block
    k_st<<<NN / 8, 256, 0, stream>>>(HW, avec, s, t);

    // 3) edge scores + column sum of squares
    k_edge_score<<<(N_E + 255) / 256, 256, 0, stream>>>(eidx, s, t, lr_e, colnorm2);

    // 4) finalize norms
    k_norm<<<(NN + 255) / 256, 256, 0, stream>>>(colnorm2);

    // 5) scatter SpMM: one wave per edge
    k_spmm<<<N_E / 8, 256, 0, stream>>>(eidx, HW, lr_e, colnorm2, out);
}